// GraphEncoder_13211319402646
// MI455X (gfx1250) — compile-verified
//
#include <hip/hip_runtime.h>

#define N_NODES 50000L
#define N_EDGES 500000L

typedef float v2f __attribute__((ext_vector_type(2)));
typedef float v8f __attribute__((ext_vector_type(8)));

// ---------------------------------------------------------------- zero fill
__global__ void zero_f32(float4* __restrict__ p, long n4) {
    long i = (long)blockIdx.x * blockDim.x + threadIdx.x;
    if (i < n4) p[i] = make_float4(0.f, 0.f, 0.f, 0.f);
}

// ------------------------------------------------- segment-sum scatter (GIN agg)
// agg[dst[e]*D + f] += x[src[e]*D + f]; one thread per (edge, 4-feature chunk).
// D/4 is a power of two -> shift/mask instead of 64-bit div.
template <int LOGDQ>   // D = (1<<LOGDQ)*4
__global__ void scatter_add_f4(const float* __restrict__ x,
                               const long long* __restrict__ src,
                               const long long* __restrict__ dst,
                               float* __restrict__ agg, long total) {
    constexpr int DQ = 1 << LOGDQ;
    constexpr int D  = DQ * 4;
    long i = (long)blockIdx.x * blockDim.x + threadIdx.x;
    if (i >= total) return;
    long e = i >> LOGDQ;
    int  j = (int)(i & (DQ - 1)) * 4;
    long s = src[e];
    long d = dst[e];
    const float4 v = *(const float4*)(x + s * (long)D + j);
    float* a = agg + d * (long)D + j;
    atomicAdd(a + 0, v.x);
    atomicAdd(a + 1, v.y);
    atomicAdd(a + 2, v.z);
    atomicAdd(a + 3, v.w);
}

// ------------------------------------------------- fused WMMA-f32 GEMM
// Out[M,N] = act( (A [+ Add]) @ W + bias ),  A: MxK row-major, W: KxN row-major.
// One wave computes a 16x64 output strip = four 16x16 WMMA tiles sharing one
// A fragment per k-step (4x arithmetic intensity on A, 4 wmma per loop body).
// grid = (M/16, N/64), block = 32 (one wave32). EXEC all-ones, no divergence.
// Compile-time K/N/flags -> pointer-increment addressing + immediate offsets.
template <int K, int N, int RELU, int HASADD>
__global__ __launch_bounds__(32)
void gemm_wmma(const float* __restrict__ A,
               const float* __restrict__ Add,     // residual (agg), used iff HASADD
               const float* __restrict__ W,
               const float* __restrict__ bias,
               float* __restrict__ Out) {
    const int lane  = threadIdx.x;        // 0..31
    const int half  = lane >> 4;          // 0: lanes 0-15, 1: lanes 16-31
    const int l15   = lane & 15;
    const long mBase = (long)blockIdx.x * 16;
    const int  nBase = blockIdx.y * 64;

    v8f acc0 = {0.f,0.f,0.f,0.f,0.f,0.f,0.f,0.f};
    v8f acc1 = acc0, acc2 = acc0, acc3 = acc0;

    // A fragment (16x4 f32): lane supplies row mBase+l15, K pair (k+2*half, +1)
    const float* __restrict__ arow   = A + (mBase + l15) * K + half * 2;
    const float* __restrict__ addrow = Add + (mBase + l15) * K + half * 2;
    // B fragments (4x16 f32): reg0 = row k+2*half, reg1 = row k+1+2*half
    const float* __restrict__ wp = W + (half * 2) * N + nBase + l15;

#pragma unroll 2
    for (int k = 0; k < K; k += 4) {
        v2f a;
        a.x = arow[0];
        a.y = arow[1];
        if (HASADD) {
            a.x += addrow[0];
            a.y += addrow[1];
            addrow += 4;
        }
        arow += 4;

        v2f b0, b1, b2, b3;
        b0.x = wp[0];      b0.y = wp[N + 0];
        b1.x = wp[16];     b1.y = wp[N + 16];
        b2.x = wp[32];     b2.y = wp[N + 32];
        b3.x = wp[48];     b3.y = wp[N + 48];
        wp += 4 * N;

        acc0 = __builtin_amdgcn_wmma_f32_16x16x4_f32(false, a, false, b0, (short)0, acc0, false, false);
        acc1 = __builtin_amdgcn_wmma_f32_16x16x4_f32(false, a, false, b1, (short)0, acc1, false, false);
        acc2 = __builtin_amdgcn_wmma_f32_16x16x4_f32(false, a, false, b2, (short)0, acc2, false, false);
        acc3 = __builtin_amdgcn_wmma_f32_16x16x4_f32(false, a, false, b3, (short)0, acc3, false, false);
    }

    const float bn0 = bias[nBase + l15 +  0];
    const float bn1 = bias[nBase + l15 + 16];
    const float bn2 = bias[nBase + l15 + 32];
    const float bn3 = bias[nBase + l15 + 48];

    // C/D layout: reg r, lanes 0-15 -> (M=r, N=lane); lanes 16-31 -> (M=r+8, N=lane-16)
    float* __restrict__ orow = Out + (mBase + half * 8) * N + nBase + l15;
#pragma unroll
    for (int r = 0; r < 8; ++r) {
        float v0 = acc0[r] + bn0;
        float v1 = acc1[r] + bn1;
        float v2 = acc2[r] + bn2;
        float v3 = acc3[r] + bn3;
        if (RELU) {
            v0 = fmaxf(v0, 0.f); v1 = fmaxf(v1, 0.f);
            v2 = fmaxf(v2, 0.f); v3 = fmaxf(v3, 0.f);
        }
        orow[(long)r * N +  0] = v0;
        orow[(long)r * N + 16] = v1;
        orow[(long)r * N + 32] = v2;
        orow[(long)r * N + 48] = v3;
    }
}

// ---------------------------------------------------------------- launcher
extern "C" void kernel_launch(void* const* d_in, const int* in_sizes, int n_in,
                              void* d_out, int out_size, void* d_ws, size_t ws_size,
                              hipStream_t stream) {
    const float*     x   = (const float*)d_in[0];
    const long long* ei  = (const long long*)d_in[1];   // (2, E) int64
    // d_in[2] = edge_attr, unused by GINConv
    const float* w1a = (const float*)d_in[3];
    const float* b1a = (const float*)d_in[4];
    const float* w1b = (const float*)d_in[5];
    const float* b1b = (const float*)d_in[6];
    const float* w2a = (const float*)d_in[7];
    const float* b2a = (const float*)d_in[8];
    const float* w2b = (const float*)d_in[9];
    const float* b2b = (const float*)d_in[10];
    const float* wl  = (const float*)d_in[11];
    const float* bl  = (const float*)d_in[12];
    float* out = (float*)d_out;
    float* ws  = (float*)d_ws;

    const long M = N_NODES;
    const long E = N_EDGES;
    const long long* src = ei;
    const long long* dst = ei + E;

    // Workspace layout (floats). h2 overlays the dead agg1/t1/h1/agg2 region
    // (16M floats >= 12.8M needed); safe due to in-stream ordering.
    float* agg1 = ws;                    // M*128
    float* t1   = agg1 + M * 128;        // M*64
    float* h1   = t1   + M * 64;         // M*64
    float* agg2 = h1   + M * 64;         // M*64
    float* t2   = agg2 + M * 64;         // M*256
    float* h2   = ws;                    // reuse front region (M*256 fits)

    const int ZB = 256;
    const unsigned MT = (unsigned)(M / 16);   // 3125 tiles, M divides exactly

    // ---- GIN layer 1 ----
    {   // agg1 = segment_sum(x[src] -> dst), D = 128
        long n4 = M * 128 / 4;
        zero_f32<<<(int)((n4 + ZB - 1) / ZB), ZB, 0, stream>>>((float4*)agg1, n4);
        long tot = E * 32;
        scatter_add_f4<5><<<(int)((tot + ZB - 1) / ZB), ZB, 0, stream>>>(
            x, src, dst, agg1, tot);
    }
    // t1 = ReLU((x + agg1) @ w1a + b1a)   [128 -> 64]
    gemm_wmma<128, 64, 1, 1><<<dim3(MT, 1), 32, 0, stream>>>(x, agg1, w1a, b1a, t1);
    // h1 = ReLU(t1 @ w1b + b1b)           [64 -> 64, fused inter-layer ReLU]
    gemm_wmma<64, 64, 1, 0><<<dim3(MT, 1), 32, 0, stream>>>(t1, nullptr, w1b, b1b, h1);

    // ---- GIN layer 2 ----
    {   // agg2 = segment_sum(h1[src] -> dst), D = 64
        long n4 = M * 64 / 4;
        zero_f32<<<(int)((n4 + ZB - 1) / ZB), ZB, 0, stream>>>((float4*)agg2, n4);
        long tot = E * 16;
        scatter_add_f4<4><<<(int)((tot + ZB - 1) / ZB), ZB, 0, stream>>>(
            h1, src, dst, agg2, tot);
    }
    // t2 = ReLU((h1 + agg2) @ w2a + b2a)  [64 -> 256]
    gemm_wmma<64, 256, 1, 1><<<dim3(MT, 4), 32, 0, stream>>>(h1, agg2, w2a, b2a, t2);
    // h2 = t2 @ w2b + b2b                 [256 -> 256, no activation]
    gemm_wmma<256, 256, 0, 0><<<dim3(MT, 4), 32, 0, stream>>>(t2, nullptr, w2b, b2b, h2);

    // ---- latent projection ----
    // out = h2 @ wl + bl                  [256 -> 128]
    gemm_wmma<256, 128, 0, 0><<<dim3(MT, 2), 32, 0, stream>>>(h2, nullptr, wl, bl, out);
}